// GCN_node_all_path_38732015075674
// MI455X (gfx1250) — compile-verified
//
#include <hip/hip_runtime.h>
#include <hip/hip_bf16.h>

// GCN_node_all_path for MI455X (gfx1250).
// Roofline: edge_feat is [512,512,128] f32 = 134MB read exactly once ->
// ~5.8us floor at 23.3 TB/s HBM; total GEMM work is only ~0.5 GFLOP over
// ~6MB of reused operands. So: (a) the reduction kernel is written for pure
// bandwidth (B128 non-temporal streaming loads, 4 independent accumulators,
// LDS tree reduce), and (b) all GEMMs use V_WMMA_F32_16X16X4_F32 -- full f32
// numerics (no precision sacrifice needed when memory-bound) while engaging
// the CDNA5 matrix pipe.

#define NN   512
#define DD   128
#define HH   4
#define LL   4
#define GHD  32
#define OUTD 128
#define OLD  256   // leading dim of per-head grown feature buffer (128 + 4*32)

typedef __attribute__((ext_vector_type(2))) float v2f;
typedef __attribute__((ext_vector_type(4))) float v4f;
typedef __attribute__((ext_vector_type(8))) float v8f;

__device__ __forceinline__ v8f wmma4(v8f c, v2f a, v2f b) {
  // D = A(16x4) * B(4x16) + C, all f32.  8-arg form:
  // (neg_a, A, neg_b, B, c_mod, C, reuse_a, reuse_b)
  return __builtin_amdgcn_wmma_f32_16x16x4_f32(
      false, a, false, b, (short)0, c, false, false);
}

// ---------------------------------------------------------------------------
// Kernel 1: edge_mean[i][d] = mean_j edge_feat[i][j][d]
// One block per node i.  256 threads = 8 waves: tid = (jgroup<<5)|dvec.
// Each thread streams a float4 column slice (32 lanes * 16B = one full 512B
// row per load wavefront, fully coalesced) with NON-TEMPORAL hint (stream is
// read-once; keep L2 free for the reused adj/activations), 4 independent
// accumulators to decouple loads from the FMA chain, then LDS tree-reduce.
// ---------------------------------------------------------------------------
__global__ void __launch_bounds__(256)
k_edge_mean(const float* __restrict__ ef, float* __restrict__ em) {
  __shared__ float s[8][DD];
  const int i   = blockIdx.x;
  const int tid = threadIdx.x;
  const int jg  = tid >> 5;          // 0..7
  const int dv  = (tid & 31) * 4;    // 0..124
  const v4f* base = reinterpret_cast<const v4f*>(
      ef + (size_t)i * NN * DD + dv);          // stride DD/4 v4f per row
  v4f a0 = {0.f, 0.f, 0.f, 0.f}, a1 = a0, a2 = a0, a3 = a0;
  // 512 rows / 8 jgroups = 64 rows per thread, 4-way unrolled (16 iters)
  for (int j = jg; j < NN; j += 32) {
    const v4f v0 = __builtin_nontemporal_load(base + (size_t)(j     ) * (DD / 4));
    const v4f v1 = __builtin_nontemporal_load(base + (size_t)(j +  8) * (DD / 4));
    const v4f v2 = __builtin_nontemporal_load(base + (size_t)(j + 16) * (DD / 4));
    const v4f v3 = __builtin_nontemporal_load(base + (size_t)(j + 24) * (DD / 4));
    a0 += v0; a1 += v1; a2 += v2; a3 += v3;
  }
  a0 += a1; a2 += a3; a0 += a2;
  *reinterpret_cast<v4f*>(&s[jg][dv]) = a0;
  __syncthreads();
  if (tid < DD) {
    float sum = 0.f;
    #pragma unroll
    for (int g = 0; g < 8; ++g) sum += s[g][tid];
    em[i * DD + tid] = sum * (1.0f / (float)NN);
  }
}

// ---------------------------------------------------------------------------
// Kernel 2: denom[h][i] = deg + (deg==0), deg = sum_j adj[h][i][j]
// One wave per (h,i) row, shuffle reduction.  adj is reused by k_layer, so
// regular-temporal loads (let it live in the 192MB L2).
// ---------------------------------------------------------------------------
__global__ void __launch_bounds__(256)
k_denom(const float* __restrict__ adj, float* __restrict__ denom) {
  const int row  = blockIdx.x * 8 + (threadIdx.x >> 5); // 0 .. H*N-1
  const int lane = threadIdx.x & 31;
  const float* a = adj + (size_t)row * NN;
  float s = 0.f;
  for (int j = lane; j < NN; j += 32) s += a[j];
  #pragma unroll
  for (int off = 16; off; off >>= 1) s += __shfl_xor(s, off, 32);
  if (lane == 0) denom[row] = s + ((s == 0.0f) ? 1.0f : 0.0f);
}

// ---------------------------------------------------------------------------
// Kernel 3: seed per-head feature buffers: outputs_h[:, 0:128] = node_feat
// ---------------------------------------------------------------------------
__global__ void __launch_bounds__(256)
k_init(const float* __restrict__ nf, float* __restrict__ outputs) {
  const int idx = blockIdx.x * 256 + threadIdx.x;   // < H*N*D
  const int h = idx >> 16;
  const int n = (idx >> 7) & (NN - 1);
  const int d = idx & (DD - 1);
  outputs[(size_t)h * NN * OLD + n * OLD + d] = nf[n * DD + d];
}

// ---------------------------------------------------------------------------
// Kernel 4: XW[h] = outputs_h[:, 0:Din] @ Wn_l[h]   ([512,Din]x[Din,32])
// grid = (M tiles 32, N tiles 2, H).  One wave per 16x16 C tile.
// Lane map: q = lane&15, kh = lane>>4.
//   A tile: lane -> row m0+q, cols k0+2*kh .. +1   (f32 16x4 layout)
//   B tile: lane -> col n0+q, rows k0+2*kh .. +1   (mirror)
//   C/D:    vgpr r, lane -> (m0 + r + 8*kh, n0+q)
// ---------------------------------------------------------------------------
__global__ void __launch_bounds__(32)
k_xw(const float* __restrict__ outputs, const float* __restrict__ Wn,
     float* __restrict__ XW, int Din) {
  const int m0 = blockIdx.x * 16, n0 = blockIdx.y * 16, h = blockIdx.z;
  const int t = threadIdx.x, q = t & 15, kh = t >> 4;
  const float* Xh = outputs + (size_t)h * NN * OLD;
  const float* Wh = Wn      + (size_t)h * Din * GHD;
  v8f c = {};
  for (int k0 = 0; k0 < Din; k0 += 4) {
    const int ka = k0 + 2 * kh;
    v2f a; a.x = Xh[(m0 + q) * OLD + ka];
           a.y = Xh[(m0 + q) * OLD + ka + 1];
    v2f b; b.x = Wh[ka * GHD + n0 + q];
           b.y = Wh[(ka + 1) * GHD + n0 + q];
    c = wmma4(c, a, b);
  }
  float* Oh = XW + (size_t)h * NN * GHD;
  #pragma unroll
  for (int r = 0; r < 8; ++r)
    Oh[(m0 + r + 8 * kh) * GHD + n0 + q] = c[r];
}

// ---------------------------------------------------------------------------
// Kernel 5: g = relu((adj_h @ XW_h + edge_mean @ W_edge[h,l]) / denom_h)
// Both GEMMs accumulate into the SAME WMMA accumulator (shared C tile).
// Writes g into outputs_h[:, 128+l*32 ..] (next layer input) and
// g + node_feat[:, l*32..] into all_heads[:, h*128 + l*32 ..] (residual).
// ---------------------------------------------------------------------------
__global__ void __launch_bounds__(32)
k_layer(const float* __restrict__ adj, const float* __restrict__ XW,
        const float* __restrict__ em, const float* __restrict__ We_all,
        const float* __restrict__ denom, const float* __restrict__ nf,
        float* __restrict__ outputs, float* __restrict__ all_heads, int l) {
  const int m0 = blockIdx.x * 16, n0 = blockIdx.y * 16, h = blockIdx.z;
  const int t = threadIdx.x, q = t & 15, kh = t >> 4;
  v8f c = {};
  // n_out contribution: adj_h [512x512] @ XW_h [512x32]
  const float* Ah = adj + (size_t)h * NN * NN;
  const float* Bh = XW  + (size_t)h * NN * GHD;
  for (int k0 = 0; k0 < NN; k0 += 4) {
    const int ka = k0 + 2 * kh;
    v2f a; a.x = Ah[(m0 + q) * NN + ka];
           a.y = Ah[(m0 + q) * NN + ka + 1];
    v2f b; b.x = Bh[ka * GHD + n0 + q];
           b.y = Bh[(ka + 1) * GHD + n0 + q];
    c = wmma4(c, a, b);
  }
  // e_out contribution: edge_mean [512x128] @ W_edge[h,l] [128x32]
  const float* We = We_all + (size_t)(h * LL + l) * DD * GHD;
  for (int k0 = 0; k0 < DD; k0 += 4) {
    const int ka = k0 + 2 * kh;
    v2f a; a.x = em[(m0 + q) * DD + ka];
           a.y = em[(m0 + q) * DD + ka + 1];
    v2f b; b.x = We[ka * GHD + n0 + q];
           b.y = We[(ka + 1) * GHD + n0 + q];
    c = wmma4(c, a, b);
  }
  // epilogue: divide by denom, ReLU, scatter both destinations
  float* Oh = outputs + (size_t)h * NN * OLD;
  #pragma unroll
  for (int r = 0; r < 8; ++r) {
    const int row = m0 + r + 8 * kh;
    const int col = n0 + q;                       // 0..31 within GHD
    float g = c[r] / denom[h * NN + row];
    g = fmaxf(g, 0.0f);
    Oh[row * OLD + DD + l * GHD + col] = g;
    all_heads[(size_t)row * (HH * DD) + h * DD + l * GHD + col] =
        g + nf[row * DD + l * GHD + col];
  }
}

// ---------------------------------------------------------------------------
// Kernel 6: out = all_heads [512x512] @ W_lin [512x128] + b_lin
// ---------------------------------------------------------------------------
__global__ void __launch_bounds__(32)
k_final(const float* __restrict__ ah, const float* __restrict__ W,
        const float* __restrict__ bias, float* __restrict__ out) {
  const int m0 = blockIdx.x * 16, n0 = blockIdx.y * 16;
  const int t = threadIdx.x, q = t & 15, kh = t >> 4;
  const int K = HH * DD;   // 512
  v8f c = {};
  for (int k0 = 0; k0 < K; k0 += 4) {
    const int ka = k0 + 2 * kh;
    v2f a; a.x = ah[(m0 + q) * K + ka];
           a.y = ah[(m0 + q) * K + ka + 1];
    v2f b; b.x = W[ka * OUTD + n0 + q];
           b.y = W[(ka + 1) * OUTD + n0 + q];
    c = wmma4(c, a, b);
  }
  const float bl = bias[n0 + q];
  #pragma unroll
  for (int r = 0; r < 8; ++r)
    out[(m0 + r + 8 * kh) * OUTD + n0 + q] = c[r] + bl;
}

// ---------------------------------------------------------------------------
extern "C" void kernel_launch(void* const* d_in, const int* in_sizes, int n_in,
                              void* d_out, int out_size, void* d_ws, size_t ws_size,
                              hipStream_t stream) {
  const float* node_feat = (const float*)d_in[0];
  const float* edge_feat = (const float*)d_in[1];
  const float* adj       = (const float*)d_in[2];
  const float* W_edge    = (const float*)d_in[3];
  const float* Wn[LL]    = {(const float*)d_in[4], (const float*)d_in[5],
                            (const float*)d_in[6], (const float*)d_in[7]};
  const float* W_lin     = (const float*)d_in[8];
  const float* b_lin     = (const float*)d_in[9];
  float* out = (float*)d_out;

  // workspace carve-up (floats): total ~3.67 MB
  float* ws        = (float*)d_ws;
  float* edge_mean = ws;                                  // N*D      = 65536
  float* denom     = edge_mean + NN * DD;                 // H*N      = 2048
  float* outputs   = denom + HH * NN;                     // H*N*OLD  = 524288
  float* XW        = outputs + (size_t)HH * NN * OLD;     // H*N*GHD  = 65536
  float* all_heads = XW + (size_t)HH * NN * GHD;          // N*H*D    = 262144

  k_edge_mean<<<NN, 256, 0, stream>>>(edge_feat, edge_mean);
  k_denom<<<(HH * NN) / 8, 256, 0, stream>>>(adj, denom);
  k_init<<<(HH * NN * DD) / 256, 256, 0, stream>>>(node_feat, outputs);

  const int Din[LL] = {DD, DD + GHD, DD + 2 * GHD, DD + 3 * GHD};
  for (int l = 0; l < LL; ++l) {
    k_xw<<<dim3(NN / 16, GHD / 16, HH), 32, 0, stream>>>(outputs, Wn[l], XW, Din[l]);
    k_layer<<<dim3(NN / 16, GHD / 16, HH), 32, 0, stream>>>(
        adj, XW, edge_mean, W_edge, denom, node_feat, outputs, all_heads, l);
  }
  k_final<<<dim3(NN / 16, OUTD / 16), 32, 0, stream>>>(all_heads, W_lin, b_lin, out);
}